// GINEncoder_10823317586227
// MI455X (gfx1250) — compile-verified
//
#include <hip/hip_runtime.h>

// GIN encoder for MI455X (gfx1250, wave32).
// Scatter phase is HBM/atomic bound (~2.5 GB traffic); GEMM phase uses
// V_WMMA_F32_16X16X4_F32 so the MLPs ride the matrix pipes in full fp32.

#define DCH 64
#define WSTRIDE 72   // LDS row stride (floats) for weight tiles: conflict-free B frags
#define TSTRIDE 68   // LDS row stride (floats) for x/h tiles: conflict-free A frags

typedef float v2f __attribute__((ext_vector_type(2)));
typedef float v8f __attribute__((ext_vector_type(8)));

__device__ __forceinline__ v8f wmma4(v2f a, v2f b, v8f c) {
  // D = A(16x4,f32) * B(4x16,f32) + C(16x16,f32)
  return __builtin_amdgcn_wmma_f32_16x16x4_f32(
      /*neg_a=*/false, a, /*neg_b=*/false, b,
      /*c_mod=*/(short)0, c, /*reuse_a=*/false, /*reuse_b=*/false);
}

// ---------------------------------------------------------------------------
// Edge scatter: one wave per edge. Lane c handles channels 2c, 2c+1.
// Coalesced 256B gather of x[src]; f32 atomic adds into agg[dst].
// ---------------------------------------------------------------------------
__global__ void __launch_bounds__(256) gin_scatter(const float2* __restrict__ x2,
                                                   const int* __restrict__ src,
                                                   const int* __restrict__ dst,
                                                   float* __restrict__ agg, int nE) {
  int gid = blockIdx.x * blockDim.x + threadIdx.x;
  int e = gid >> 5;
  if (e >= nE) return;
  int lane = gid & 31;
  int s = src[e];
  int d = dst[e];
  float2 v = x2[s * 32 + lane];
  atomicAdd(&agg[d * 64 + lane * 2 + 0], v.x);
  atomicAdd(&agg[d * 64 + lane * 2 + 1], v.y);
}

// ---------------------------------------------------------------------------
// Fragment helpers (ISA 7.12.2 layouts, wave32).
// A (16x4 f32): lane -> M=lane&15; v0 holds K = k0 + 2*(lane>=16), v1 = K+1.
// B (4x16 f32): lane -> N=lane&15; v0 holds K = k0 + 2*(lane>=16), v1 = K+1.
// C (16x16 f32): VGPR r, lane -> M = r + 8*(lane>=16), N = lane&15.
// ---------------------------------------------------------------------------
__device__ __forceinline__ v2f load_a_frag(const float* tile, int l15, int hi, int k0) {
  v2f a;
  a.x = tile[l15 * TSTRIDE + k0 + 2 * hi];
  a.y = tile[l15 * TSTRIDE + k0 + 1 + 2 * hi];
  return a;
}
__device__ __forceinline__ v2f load_b_frag(const float* sW, int l15, int hi, int k0, int n0) {
  v2f b;
  b.x = sW[(k0 + 2 * hi) * WSTRIDE + n0 + l15];
  b.y = sW[(k0 + 1 + 2 * hi) * WSTRIDE + n0 + l15];
  return b;
}

// ---------------------------------------------------------------------------
// Fused GIN MLP: xout = relu(relu(((1+eps)*x + agg) @ W1 + b1) @ W2 + b2)
// 8 waves/block, each wave owns a 16-node tile; W1/W2 staged in LDS.
// ---------------------------------------------------------------------------
__global__ void __launch_bounds__(256) gin_mlp(const float* __restrict__ xin,
                                               const float* __restrict__ agg,
                                               const float* __restrict__ W1,
                                               const float* __restrict__ b1,
                                               const float* __restrict__ W2,
                                               const float* __restrict__ b2,
                                               const float* __restrict__ epsp,
                                               float* __restrict__ xout, int nNodes) {
  __shared__ float sW1[DCH * WSTRIDE];
  __shared__ float sW2[DCH * WSTRIDE];
  __shared__ float sB1[DCH];
  __shared__ float sB2[DCH];
  __shared__ float tiles[8][16 * TSTRIDE];

  for (int i = threadIdx.x; i < DCH * DCH; i += 256) {
    int k = i >> 6, n = i & 63;
    sW1[k * WSTRIDE + n] = W1[i];
    sW2[k * WSTRIDE + n] = W2[i];
  }
  if (threadIdx.x < DCH) {
    sB1[threadIdx.x] = b1[threadIdx.x];
    sB2[threadIdx.x] = b2[threadIdx.x];
  }
  __syncthreads();

  const int wave = threadIdx.x >> 5;
  const int lane = threadIdx.x & 31;
  const int l15 = lane & 15;
  const int hi = lane >> 4;
  const int base = blockIdx.x * 128 + wave * 16;
  if (base >= nNodes) return;

  float* tile = tiles[wave];
  const float e1 = 1.0f + epsp[0];

  // Stage (1+eps)*x + agg into LDS tile [16][TSTRIDE] (float4, coalesced).
  const float4* x4 = (const float4*)xin;
  const float4* a4 = (const float4*)agg;
#pragma unroll
  for (int i = 0; i < 8; ++i) {
    int f = lane + 32 * i;   // float4 index in 16x16-float4 tile
    int row = f >> 4;
    int c4 = f & 15;
    float4 xv = x4[(size_t)(base + row) * 16 + c4];
    float4 av = a4[(size_t)(base + row) * 16 + c4];
    float4 r;
    r.x = e1 * xv.x + av.x;
    r.y = e1 * xv.y + av.y;
    r.z = e1 * xv.z + av.z;
    r.w = e1 * xv.w + av.w;
    *(float4*)&tile[row * TSTRIDE + c4 * 4] = r;
  }

  // GEMM1: h = tile(16x64) @ W1(64x64)
  v8f acc[4];
#pragma unroll
  for (int t = 0; t < 4; ++t) acc[t] = (v8f){};
  for (int k0 = 0; k0 < DCH; k0 += 4) {
    v2f a = load_a_frag(tile, l15, hi, k0);
#pragma unroll
    for (int t = 0; t < 4; ++t)
      acc[t] = wmma4(a, load_b_frag(sW1, l15, hi, k0, t * 16), acc[t]);
  }

  // bias + relu, re-stripe h into the (now dead) input tile. Same-wave DS ops
  // are in-order, so no barrier needed for this private region.
#pragma unroll
  for (int t = 0; t < 4; ++t) {
    float bias = sB1[t * 16 + l15];
#pragma unroll
    for (int r = 0; r < 8; ++r)
      tile[(r + 8 * hi) * TSTRIDE + t * 16 + l15] = fmaxf(acc[t][r] + bias, 0.0f);
  }

  // GEMM2: x' = h(16x64) @ W2(64x64)
#pragma unroll
  for (int t = 0; t < 4; ++t) acc[t] = (v8f){};
  for (int k0 = 0; k0 < DCH; k0 += 4) {
    v2f a = load_a_frag(tile, l15, hi, k0);
#pragma unroll
    for (int t = 0; t < 4; ++t)
      acc[t] = wmma4(a, load_b_frag(sW2, l15, hi, k0, t * 16), acc[t]);
  }

  // bias + relu, store to global.
#pragma unroll
  for (int t = 0; t < 4; ++t) {
    float bias = sB2[t * 16 + l15];
#pragma unroll
    for (int r = 0; r < 8; ++r) {
      float v = fmaxf(acc[t][r] + bias, 0.0f);
      xout[(size_t)(base + r + 8 * hi) * DCH + t * 16 + l15] = v;
    }
  }
}

// ---------------------------------------------------------------------------
// Final linear: out = x @ Wf + bf  (no relu)
// ---------------------------------------------------------------------------
__global__ void __launch_bounds__(256) gin_final(const float* __restrict__ xin,
                                                 const float* __restrict__ Wf,
                                                 const float* __restrict__ bf,
                                                 float* __restrict__ out, int nNodes) {
  __shared__ float sW[DCH * WSTRIDE];
  __shared__ float sB[DCH];
  __shared__ float tiles[8][16 * TSTRIDE];

  for (int i = threadIdx.x; i < DCH * DCH; i += 256) {
    int k = i >> 6, n = i & 63;
    sW[k * WSTRIDE + n] = Wf[i];
  }
  if (threadIdx.x < DCH) sB[threadIdx.x] = bf[threadIdx.x];
  __syncthreads();

  const int wave = threadIdx.x >> 5;
  const int lane = threadIdx.x & 31;
  const int l15 = lane & 15;
  const int hi = lane >> 4;
  const int base = blockIdx.x * 128 + wave * 16;
  if (base >= nNodes) return;

  float* tile = tiles[wave];
  const float4* x4 = (const float4*)xin;
#pragma unroll
  for (int i = 0; i < 8; ++i) {
    int f = lane + 32 * i;
    int row = f >> 4;
    int c4 = f & 15;
    *(float4*)&tile[row * TSTRIDE + c4 * 4] = x4[(size_t)(base + row) * 16 + c4];
  }

  v8f acc[4];
#pragma unroll
  for (int t = 0; t < 4; ++t) acc[t] = (v8f){};
  for (int k0 = 0; k0 < DCH; k0 += 4) {
    v2f a = load_a_frag(tile, l15, hi, k0);
#pragma unroll
    for (int t = 0; t < 4; ++t)
      acc[t] = wmma4(a, load_b_frag(sW, l15, hi, k0, t * 16), acc[t]);
  }

#pragma unroll
  for (int t = 0; t < 4; ++t) {
    float bias = sB[t * 16 + l15];
#pragma unroll
    for (int r = 0; r < 8; ++r)
      out[(size_t)(base + r + 8 * hi) * DCH + t * 16 + l15] = acc[t][r] + bias;
  }
}

// ---------------------------------------------------------------------------
extern "C" void kernel_launch(void* const* d_in, const int* in_sizes, int n_in,
                              void* d_out, int out_size, void* d_ws, size_t ws_size,
                              hipStream_t stream) {
  const float* x   = (const float*)d_in[0];
  const int*   ei  = (const int*)d_in[1];
  const float* W1  = (const float*)d_in[2];
  const float* b1  = (const float*)d_in[3];
  const float* W2  = (const float*)d_in[4];
  const float* b2  = (const float*)d_in[5];
  const float* eps = (const float*)d_in[6];
  const float* Wf  = (const float*)d_in[7];
  const float* bf  = (const float*)d_in[8];
  float* out = (float*)d_out;

  const int N = in_sizes[0] / DCH;   // 100000
  const int E = in_sizes[1] / 2;     // 1600000
  const int L = in_sizes[6];         // 3

  float* agg  = (float*)d_ws;
  float* buf0 = agg + (size_t)N * DCH;
  float* buf1 = buf0 + (size_t)N * DCH;

  const int mlpBlocks  = (N + 127) / 128;
  const int scatBlocks = (E + 7) / 8;   // one wave32 per edge, 8 waves/block

  const float* cur = x;
  for (int l = 0; l < L; ++l) {
    hipMemsetAsync(agg, 0, (size_t)N * DCH * sizeof(float), stream);
    gin_scatter<<<scatBlocks, 256, 0, stream>>>((const float2*)cur, ei, ei + E, agg, E);
    float* nxt = (l & 1) ? buf1 : buf0;
    gin_mlp<<<mlpBlocks, 256, 0, stream>>>(cur, agg,
                                           W1 + (size_t)l * DCH * DCH, b1 + (size_t)l * DCH,
                                           W2 + (size_t)l * DCH * DCH, b2 + (size_t)l * DCH,
                                           eps + l, nxt, N);
    cur = nxt;
  }
  gin_final<<<mlpBlocks, 256, 0, stream>>>(cur, Wf, bf, out, N);
}